// GraphPolicyNetwork_36335423324413
// MI455X (gfx1250) — compile-verified
//
#include <hip/hip_runtime.h>
#include <cstdint>

typedef float v2f __attribute__((ext_vector_type(2)));
typedef float v8f __attribute__((ext_vector_type(8)));

// ---------------------------------------------------------------------------
// Utility kernels
// ---------------------------------------------------------------------------
__global__ void fill_f32(float* __restrict__ p, float v, long long n) {
    long long i = (long long)blockIdx.x * blockDim.x + threadIdx.x;
    if (i < n) p[i] = v;
}

__global__ void deg_count(const long long* __restrict__ dst, float* __restrict__ deg, int nE) {
    int e = blockIdx.x * blockDim.x + threadIdx.x;
    if (e < nE) atomicAdd(&deg[(int)dst[e]], 1.0f);
}

__global__ void rsqrt_inplace(float* __restrict__ d, int n) {
    int i = blockIdx.x * blockDim.x + threadIdx.x;
    if (i < n) {
        float v = d[i];
        d[i] = (v > 0.0f) ? __frsqrt_rn(v) : 0.0f;
    }
}

// ---------------------------------------------------------------------------
// Dense GEMM via V_WMMA_F32_16X16X4_F32.
//   H[M,NCOLS] = A[M,128] @ W[128,NCOLS]
// One wave (32 lanes) computes a 16-row stripe across all NCOLS, reusing the
// A fragment across the N-tiles. Layouts per CDNA5 ISA:
//   A 16x4:  lanes 0-15 -> M=lane, {v0,v1}={K,K+1}; lanes 16-31 -> {K+2,K+3}
//   B 4x16:  lanes 0-15 -> N=lane, {v0,v1}={K,K+1}; lanes 16-31 -> {K+2,K+3}
//   C/D:     VGPR r: lanes 0-15 M=r, lanes 16-31 M=r+8; N = lane&15
// ---------------------------------------------------------------------------
template <int NCOLS>
__global__ void gemm_wmma_f32(const float* __restrict__ A,
                              const float* __restrict__ W,
                              float* __restrict__ H,
                              int M) {
    constexpr int K  = 128;
    constexpr int NT = NCOLS / 16;

    const int m0   = blockIdx.x * 16;          // one wave per block (blockDim=32)
    if (m0 >= M) return;                        // wave-uniform: EXEC stays all-1s
    const int lane = threadIdx.x;
    const int half = lane >> 4;                 // 0 or 1
    const int l    = lane & 15;

    v8f acc[NT];
    const v8f vzero = {0.f, 0.f, 0.f, 0.f, 0.f, 0.f, 0.f, 0.f};
#pragma unroll
    for (int t = 0; t < NT; ++t) acc[t] = vzero;

    const float* arow = A + (size_t)(m0 + l) * K;

    for (int k = 0; k < K; k += 4) {
        const int ka = k + half * 2;
        v2f af;
        af.x = arow[ka];
        af.y = arow[ka + 1];
#pragma unroll
        for (int t = 0; t < NT; ++t) {
            const int n = t * 16 + l;
            v2f bf;
            bf.x = W[(size_t)ka * NCOLS + n];
            bf.y = W[(size_t)(ka + 1) * NCOLS + n];
            acc[t] = __builtin_amdgcn_wmma_f32_16x16x4_f32(
                /*neg_a=*/false, af, /*neg_b=*/false, bf,
                /*c_mod=*/(short)0, acc[t], /*reuse_a=*/false, /*reuse_b=*/false);
        }
    }

#pragma unroll
    for (int t = 0; t < NT; ++t) {
        const int n = t * 16 + l;
#pragma unroll
        for (int r = 0; r < 8; ++r) {
            const int m = m0 + r + half * 8;
            H[(size_t)m * NCOLS + n] = acc[t][r];
        }
    }
}

// ---------------------------------------------------------------------------
// Edge message-passing: Agg[dst] += dinv[src]*dinv[dst] * H[src]
// C/4 lanes per edge; float4 gather + 4x global_atomic_add_f32 scatter.
// ---------------------------------------------------------------------------
template <int C>
__global__ void scatter_edges(const float* __restrict__ H,
                              const long long* __restrict__ src,
                              const long long* __restrict__ dst,
                              const float* __restrict__ dinv,
                              float* __restrict__ Agg,
                              int nE) {
    constexpr int LPE = C / 4;  // threads per edge
    const long long gid = (long long)blockIdx.x * blockDim.x + threadIdx.x;
    const long long e   = gid / LPE;
    const int       q   = (int)(gid % LPE) * 4;
    if (e >= nE) return;

    const int s = (int)src[e];
    const int d = (int)dst[e];
    const float nrm = dinv[s] * dinv[d];

    const float4 v = *(const float4*)(H + (size_t)s * C + q);
    float* out = Agg + (size_t)d * C + q;
    atomicAdd(out + 0, nrm * v.x);
    atomicAdd(out + 1, nrm * v.y);
    atomicAdd(out + 2, nrm * v.z);
    atomicAdd(out + 3, nrm * v.w);
}

// ---------------------------------------------------------------------------
// Layer-1 finalize (in place on Agg): Agg = relu(Agg + dinv^2 * Hlin + b)
// ---------------------------------------------------------------------------
__global__ void finalize_relu(float* __restrict__ Agg,
                              const float* __restrict__ Hlin,
                              const float* __restrict__ dinv,
                              const float* __restrict__ bias,
                              int nNodes, int C) {
    const long long i = (long long)blockIdx.x * blockDim.x + threadIdx.x;
    if (i >= (long long)nNodes * C) return;
    const int node = (int)(i / C);
    const int c    = (int)(i % C);
    const float di = dinv[node];
    const float val = Agg[i] + di * di * Hlin[i] + bias[c];
    Agg[i] = fmaxf(val, 0.0f);
}

// ---------------------------------------------------------------------------
// Layer-2 finalize + softmax over 64 channels. One wave per node, 2 ch/lane;
// wave32 shuffle reductions for max and sum.
// ---------------------------------------------------------------------------
__global__ void finalize_softmax64(const float* __restrict__ Agg,
                                   const float* __restrict__ Hlin,
                                   const float* __restrict__ dinv,
                                   const float* __restrict__ bias,
                                   float* __restrict__ out,
                                   int nNodes) {
    const int node = blockIdx.x * (blockDim.x >> 5) + (threadIdx.x >> 5);
    const int lane = threadIdx.x & 31;
    if (node >= nNodes) return;

    const float di = dinv[node];
    const float sl = di * di;
    const size_t base = (size_t)node * 64;

    float x0 = Agg[base + lane]      + sl * Hlin[base + lane]      + bias[lane];
    float x1 = Agg[base + 32 + lane] + sl * Hlin[base + 32 + lane] + bias[32 + lane];

    float m = fmaxf(x0, x1);
#pragma unroll
    for (int off = 16; off > 0; off >>= 1) m = fmaxf(m, __shfl_xor(m, off, 32));

    const float e0 = __expf(x0 - m);
    const float e1 = __expf(x1 - m);
    float sum = e0 + e1;
#pragma unroll
    for (int off = 16; off > 0; off >>= 1) sum += __shfl_xor(sum, off, 32);

    const float inv = 1.0f / sum;
    out[base + lane]      = e0 * inv;
    out[base + 32 + lane] = e1 * inv;
}

// ---------------------------------------------------------------------------
// Host launcher
// ---------------------------------------------------------------------------
extern "C" void kernel_launch(void* const* d_in, const int* in_sizes, int n_in,
                              void* d_out, int out_size, void* d_ws, size_t ws_size,
                              hipStream_t stream) {
    (void)n_in; (void)out_size; (void)ws_size;

    const int IN_CH  = 128;
    const int HID_CH = 128;
    const int OUT_CH = 64;

    const float*     x   = (const float*)d_in[0];
    const long long* ei  = (const long long*)d_in[1];   // int64 per reference
    const float*     W1  = (const float*)d_in[2];
    const float*     b1  = (const float*)d_in[3];
    const float*     W2  = (const float*)d_in[4];
    const float*     b2  = (const float*)d_in[5];
    float*           out = (float*)d_out;

    const int N = in_sizes[0] / IN_CH;   // 50000
    const int E = in_sizes[1] / 2;       // 800000

    const long long* src = ei;
    const long long* dst = ei + E;

    // Workspace layout (floats): H1[N*128] | A1[N*128] | H2[N*64] | A2[N*64] | dinv[N]
    float* H1  = (float*)d_ws;
    float* A1  = H1 + (size_t)N * HID_CH;
    float* H2  = A1 + (size_t)N * HID_CH;
    float* A2  = H2 + (size_t)N * OUT_CH;
    float* dnv = A2 + (size_t)N * OUT_CH;

    const int TB = 256;
    auto cdiv = [](long long a, long long b) { return (int)((a + b - 1) / b); };

    // --- degree (self-loops counted via init=1.0) and D^-1/2 ---
    fill_f32<<<cdiv(N, TB), TB, 0, stream>>>(dnv, 1.0f, N);
    deg_count<<<cdiv(E, TB), TB, 0, stream>>>(dst, dnv, E);
    rsqrt_inplace<<<cdiv(N, TB), TB, 0, stream>>>(dnv, N);

    // --- layer 1: H1 = x @ W1 ---
    gemm_wmma_f32<HID_CH><<<N / 16, 32, 0, stream>>>(x, W1, H1, N);

    fill_f32<<<cdiv((long long)N * HID_CH, TB), TB, 0, stream>>>(A1, 0.0f, (long long)N * HID_CH);
    scatter_edges<HID_CH><<<cdiv((long long)E * (HID_CH / 4), TB), TB, 0, stream>>>(
        H1, src, dst, dnv, A1, E);
    finalize_relu<<<cdiv((long long)N * HID_CH, TB), TB, 0, stream>>>(
        A1, H1, dnv, b1, N, HID_CH);   // A1 now holds relu'd hidden activations

    // --- layer 2: H2 = A1 @ W2 ---
    gemm_wmma_f32<OUT_CH><<<N / 16, 32, 0, stream>>>(A1, W2, H2, N);

    fill_f32<<<cdiv((long long)N * OUT_CH, TB), TB, 0, stream>>>(A2, 0.0f, (long long)N * OUT_CH);
    scatter_edges<OUT_CH><<<cdiv((long long)E * (OUT_CH / 4), TB), TB, 0, stream>>>(
        H2, src, dst, dnv, A2, E);

    // --- finalize + softmax -> d_out (8 nodes per 256-thread block) ---
    finalize_softmax64<<<cdiv(N, 8), TB, 0, stream>>>(A2, H2, dnv, b2, out, N);
}